// AttentionDecoderRNN_66700842106907
// MI455X (gfx1250) — compile-verified
//
#include <hip/hip_runtime.h>
#include <math.h>

// ---------------------------------------------------------------------------
// AttentionDecoderRNN step, MI455X (gfx1250, wave32).
// Memory-bound GEMV pipeline (~760 MB fp32 weights -> ~33 us @ 23.3 TB/s).
// Big GEMVs run through V_WMMA_F32_16X16X4_F32: 16 output rows per wave,
// K consumed 8 at a time via b128 loads feeding two WMMAs (K-permutation
// applied identically to A and B, so the dot product is unchanged).
// ---------------------------------------------------------------------------

#define H_   2048
#define OUT_ 32000
#define S_   2048

typedef float v2f __attribute__((ext_vector_type(2)));
typedef float v4f __attribute__((ext_vector_type(4)));
typedef float v8f __attribute__((ext_vector_type(8)));

__device__ __forceinline__ v8f wmma_acc(v2f a, v2f b, v8f c) {
    // (neg_a, A, neg_b, B, c_mod, C, reuse_a, reuse_b)
    return __builtin_amdgcn_wmma_f32_16x16x4_f32(false, a, false, b, (short)0, c,
                                                 false, false);
}

// One K-chunk of 8: each lane does a single b128 load of W row data and a
// b128 load of the x chunk, split into two WMMAs. Lane half selects which
// 4 physical k's this lane's A/B slots carry; A and B agree, so D
// accumulates sum_k W[r,k]*x[k] over the full chunk.
__device__ __forceinline__ v8f gemv_step8(const float* __restrict__ wrow,
                                          const float* __restrict__ xp,
                                          int k0, int half, v8f acc) {
    const int off = k0 + 4 * half;
    v4f a4 = *(const v4f*)(wrow + off);
    v4f b4 = *(const v4f*)(xp + off);
    v2f a01 = {a4.x, a4.y}, b01 = {b4.x, b4.y};
    v2f a23 = {a4.z, a4.w}, b23 = {b4.z, b4.w};
    acc = wmma_acc(a01, b01, acc);
    acc = wmma_acc(a23, b23, acc);
    return acc;
}

// y[r] = dot(W[r,:], x) + bias[r]; 16 rows per wave, 8 waves (128 rows)/block.
// A-tile: lane m (0..15) holds W row (r0+m). B-tile: x chunk broadcast across
// all 16 N columns, so every D column carries the identical 16 results;
// lanes 0/16 extract rows 0..7 / 8..15 from the 8-VGPR accumulator.
__global__ void gemv_wmma(const float* __restrict__ W,
                          const float* __restrict__ x,
                          const float* __restrict__ bias,
                          float* __restrict__ y, int K) {
    const int lane = threadIdx.x & 31;
    const int wave = threadIdx.x >> 5;
    const int m    = lane & 15;
    const int half = lane >> 4;
    const int r0   = blockIdx.x * 128 + wave * 16;

    const float* wrow = W + (size_t)(r0 + m) * (size_t)K;

    v8f acc = {};
    for (int k0 = 0; k0 < K; k0 += 8) {
        if ((k0 & 56) == 0)  // uniform guard: EXEC stays all-1s at WMMA
            __builtin_prefetch(wrow + k0 + 512, 0, 0);
        acc = gemv_step8(wrow, x, k0, half, acc);
    }
    if (m == 0) {                       // lane 0 -> rows 0..7, lane 16 -> 8..15
        const int base = r0 + 8 * half;
        #pragma unroll
        for (int j = 0; j < 8; ++j) {
            float v = acc[j];
            if (bias) v += bias[base + j];
            y[base + j] = v;
        }
    }
}

// gates[r] = W_ih[r,:]·xcat + W_hh[r,:]·h + b_ih[r] + b_hh[r]
__global__ void gates_gemv_wmma(const float* __restrict__ Wih,
                                const float* __restrict__ Whh,
                                const float* __restrict__ xcat,
                                const float* __restrict__ h,
                                const float* __restrict__ bih,
                                const float* __restrict__ bhh,
                                float* __restrict__ gates) {
    const int lane = threadIdx.x & 31;
    const int wave = threadIdx.x >> 5;
    const int m    = lane & 15;
    const int half = lane >> 4;
    const int r0   = blockIdx.x * 128 + wave * 16;

    v8f acc = {};
    {
        const float* wrow = Wih + (size_t)(r0 + m) * (size_t)(2 * H_);
        for (int k0 = 0; k0 < 2 * H_; k0 += 8) {
            if ((k0 & 56) == 0) __builtin_prefetch(wrow + k0 + 512, 0, 0);
            acc = gemv_step8(wrow, xcat, k0, half, acc);
        }
    }
    {
        const float* wrow = Whh + (size_t)(r0 + m) * (size_t)H_;
        for (int k0 = 0; k0 < H_; k0 += 8) {
            if ((k0 & 56) == 0) __builtin_prefetch(wrow + k0 + 512, 0, 0);
            acc = gemv_step8(wrow, h, k0, half, acc);
        }
    }
    if (m == 0) {
        const int base = r0 + 8 * half;
        #pragma unroll
        for (int j = 0; j < 8; ++j)
            gates[base + j] = acc[j] + bih[base + j] + bhh[base + j];
    }
}

__global__ void build_xcat(const float* __restrict__ a,
                           const float* __restrict__ b,
                           float* __restrict__ xcat) {
    int i = blockIdx.x * 256 + threadIdx.x;
    if (i < H_)           xcat[i] = a[i];
    else if (i < 2 * H_)  xcat[i] = b[i - H_];
}

__global__ void lstm_pointwise(const float* __restrict__ gates,
                               const float* __restrict__ c0,
                               float* __restrict__ h_ws,
                               float* __restrict__ out_h,
                               float* __restrict__ out_c) {
    int j = blockIdx.x * 256 + threadIdx.x;
    if (j >= H_) return;
    float ig = 1.f / (1.f + expf(-gates[j]));
    float fg = 1.f / (1.f + expf(-gates[H_ + j]));
    float gg = tanhf(gates[2 * H_ + j]);
    float og = 1.f / (1.f + expf(-gates[3 * H_ + j]));
    float c  = fg * c0[j] + ig * gg;
    float hn = og * tanhf(c);
    h_ws[j]  = hn;
    out_h[j] = hn;
    out_c[j] = c;
}

// t[i] = sum_j h[j] * W_a[j*H + i]   (coalesced column sweep)
__global__ void wa_t_kernel(const float* __restrict__ Wa,
                            const float* __restrict__ h,
                            float* __restrict__ t) {
    int i = blockIdx.x * 256 + threadIdx.x;
    if (i >= H_) return;
    float s = 0.f;
    for (int j = 0; j < H_; ++j) s += h[j] * Wa[(size_t)j * H_ + i];
    t[i] = s;
}

// softmax over 2048 energies, single block of 1024 threads, 2 elems/thread.
// (b_a·h shift dropped: softmax is invariant to constant offsets.)
__global__ void softmax2048(const float* __restrict__ e,
                            float* __restrict__ attn_ws,
                            float* __restrict__ attn_out) {
    __shared__ float red[1024];
    int t = threadIdx.x;
    float a = e[t], b = e[t + 1024];
    red[t] = fmaxf(a, b);
    __syncthreads();
    for (int s = 512; s > 0; s >>= 1) {
        if (t < s) red[t] = fmaxf(red[t], red[t + s]);
        __syncthreads();
    }
    float M = red[0];
    __syncthreads();
    float ea = expf(a - M), eb = expf(b - M);
    red[t] = ea + eb;
    __syncthreads();
    for (int s = 512; s > 0; s >>= 1) {
        if (t < s) red[t] += red[t + s];
        __syncthreads();
    }
    float inv = 1.f / red[0];
    attn_ws[t] = ea * inv;  attn_ws[t + 1024] = eb * inv;
    attn_out[t] = ea * inv; attn_out[t + 1024] = eb * inv;
}

// ctx[i] = sum_s attn[s] * enc[s*H + i]
__global__ void context_kernel(const float* __restrict__ attn,
                               const float* __restrict__ enc,
                               float* __restrict__ ctx_ws,
                               float* __restrict__ ctx_out) {
    int i = blockIdx.x * 256 + threadIdx.x;
    if (i >= H_) return;
    float s = 0.f;
    for (int sx = 0; sx < S_; ++sx) s += attn[sx] * enc[(size_t)sx * H_ + i];
    ctx_ws[i]  = s;
    ctx_out[i] = s;
}

extern "C" void kernel_launch(void* const* d_in, const int* in_sizes, int n_in,
                              void* d_out, int out_size, void* d_ws, size_t ws_size,
                              hipStream_t stream) {
    const float* input   = (const float*)d_in[0];
    const float* lastctx = (const float*)d_in[1];
    const float* h0      = (const float*)d_in[2];
    const float* c0      = (const float*)d_in[3];
    const float* enc     = (const float*)d_in[4];
    const float* W_a     = (const float*)d_in[5];
    /* d_in[6] = b_a: dropped (softmax shift-invariant) */
    const float* W_ih    = (const float*)d_in[7];
    const float* W_hh    = (const float*)d_in[8];
    const float* b_ih    = (const float*)d_in[9];
    const float* b_hh    = (const float*)d_in[10];
    const float* W_out   = (const float*)d_in[11];
    const float* b_out   = (const float*)d_in[12];

    float* out      = (float*)d_out;
    float* o_output = out;            // 32000
    float* o_ctx    = out + 32000;    // 2048
    float* o_h      = out + 34048;    // 2048
    float* o_c      = out + 36096;    // 2048
    float* o_attn   = out + 38144;    // 2048

    float* ws      = (float*)d_ws;
    float* w_hnew  = ws;              // [0,2048)  } contiguous -> [h_new, ctx]
    float* w_ctx   = ws + 2048;       // [2048,4096)  is x for W_out GEMV
    float* w_xcat  = ws + 4096;       // 4096
    float* w_gates = ws + 8192;       // 8192
    float* w_t     = ws + 16384;      // 2048
    float* w_e     = ws + 18432;      // 2048
    float* w_attn  = ws + 20480;      // 2048

    build_xcat<<<16, 256, 0, stream>>>(input, lastctx, w_xcat);
    gates_gemv_wmma<<<64, 256, 0, stream>>>(W_ih, W_hh, w_xcat, h0,
                                            b_ih, b_hh, w_gates);
    lstm_pointwise<<<8, 256, 0, stream>>>(w_gates, c0, w_hnew, o_h, o_c);
    wa_t_kernel<<<8, 256, 0, stream>>>(W_a, w_hnew, w_t);
    gemv_wmma<<<16, 256, 0, stream>>>(enc, w_t, nullptr, w_e, 2048);      // energies
    softmax2048<<<1, 1024, 0, stream>>>(w_e, w_attn, o_attn);
    context_kernel<<<8, 256, 0, stream>>>(w_attn, enc, w_ctx, o_ctx);
    gemv_wmma<<<250, 256, 0, stream>>>(W_out, w_hnew, b_out, o_output, 4096);
}